// GraphModule_68066641707590
// MI455X (gfx1250) — compile-verified
//
#include <hip/hip_runtime.h>
#include <math.h>

typedef float v2f __attribute__((ext_vector_type(2)));
typedef float v8f __attribute__((ext_vector_type(8)));

#define N_NODES 100000
#define N_EDGES 1600000
#define D 64

// ---------------------------------------------------------------------------
// Kernel 1: zero the h_new accumulator in workspace (float4 stores).
// ---------------------------------------------------------------------------
__global__ void zero_acc_kernel(float4* __restrict__ p, int n4) {
    int i = blockIdx.x * blockDim.x + threadIdx.x;
    if (i < n4) {
        float4 z; z.x = 0.f; z.y = 0.f; z.z = 0.f; z.w = 0.f;
        p[i] = z;
    }
}

// ---------------------------------------------------------------------------
// Kernel 2: edge scatter.  msg_e = h[src_e] * w_e, atomically accumulated into
// acc[dst_e].  16 threads per edge, 4 features each (float4 gather from h,
// 4x global_atomic_add_f32 no-return into L2-resident acc).
// ---------------------------------------------------------------------------
__global__ void edge_scatter_kernel(const float* __restrict__ h,
                                    const float* __restrict__ w,
                                    const int*   __restrict__ src,
                                    const int*   __restrict__ dst,
                                    float*       __restrict__ acc) {
    long long tid  = (long long)blockIdx.x * blockDim.x + threadIdx.x;
    int       e    = (int)(tid >> 4);
    int       part = (int)(tid & 15);
    if (e >= N_EDGES) return;

    int   s  = src[e];
    int   d  = dst[e];
    float we = w[e];

    const float4 hv = *(const float4*)(h + (long long)s * D + part * 4);
    float* out = acc + (long long)d * D + part * 4;
    atomicAdd(out + 0, hv.x * we);
    atomicAdd(out + 1, hv.y * we);
    atomicAdd(out + 2, hv.z * we);
    atomicAdd(out + 3, hv.w * we);
}

// ---------------------------------------------------------------------------
// Kernel 3: fused dual-GEMM GRU cell using V_WMMA_F32_16X16X4_F32.
//   gi = x @ W_ih.T ; gh = h @ W_hh.T  (x = aggregated messages)
// One wave owns a 16-row x 16-col output block and carries 6 fp32 accumulator
// tiles (r,z,n columns of both GEMMs) across the K=64 reduction (16 steps of
// K=4, 6 WMMAs per step = 96 WMMAs/wave).
//
// Fragment layouts (ISA 7.12.2, wave32):
//   A 16x4 f32 : lane L holds row m = L&15, float2 = K { k0+2*(L>>4), +1 }
//   B 4x16 f32 : lane L holds col n = L&15, float2 = K { k0+2*(L>>4), +1 }
//   C/D 16x16  : VGPR i, lane L -> row (i + 8*(L>>4)), col (L&15)
// ---------------------------------------------------------------------------
__device__ __forceinline__ v8f wmma_f32(v2f a, v2f b, v8f c) {
    return __builtin_amdgcn_wmma_f32_16x16x4_f32(
        /*neg_a=*/false, a, /*neg_b=*/false, b,
        /*c_mod=*/(short)0, c, /*reuse_a=*/false, /*reuse_b=*/false);
}

__device__ __forceinline__ float sigmoid_f(float v) {
    return 1.0f / (1.0f + __expf(-v));
}

__global__ void __launch_bounds__(256)
gru_wmma_kernel(const float* __restrict__ x,    // h_new aggregate [N,64]
                const float* __restrict__ h,    // previous h      [N,64]
                const float* __restrict__ Wih,  // [192,64]
                const float* __restrict__ Whh,  // [192,64]
                const float* __restrict__ bih,  // [192]
                const float* __restrict__ bhh,  // [192]
                float*       __restrict__ out)  // [N,64]
{
    const int wave = (int)((blockIdx.x * blockDim.x + threadIdx.x) >> 5);
    const int lane = threadIdx.x & 31;
    const int rb   = wave >> 2;                 // row block   0..6249
    const int fb   = wave & 3;                  // feat block  0..3
    if (rb >= N_NODES / 16) return;

    const int row0 = rb * 16;
    const int c    = fb * 16;
    const int m16  = lane & 15;                 // row (A) / col (B,C,D)
    const int kh   = lane >> 4;                 // K-half select

    const float* pax = x   + (long long)(row0 + m16) * D + 2 * kh;
    const float* pah = h   + (long long)(row0 + m16) * D + 2 * kh;
    const float* pir = Wih + (      c + m16) * D + 2 * kh;
    const float* piz = Wih + ( 64 + c + m16) * D + 2 * kh;
    const float* pin = Wih + (128 + c + m16) * D + 2 * kh;
    const float* phr = Whh + (      c + m16) * D + 2 * kh;
    const float* phz = Whh + ( 64 + c + m16) * D + 2 * kh;
    const float* phn = Whh + (128 + c + m16) * D + 2 * kh;

    v8f air = {}, aiz = {}, ain = {};
    v8f ahr = {}, ahz = {}, ahn = {};

#pragma unroll
    for (int k0 = 0; k0 < D; k0 += 4) {
        v2f ax = *(const v2f*)(pax + k0);
        v2f ah = *(const v2f*)(pah + k0);
        v2f br = *(const v2f*)(pir + k0);
        v2f bz = *(const v2f*)(piz + k0);
        v2f bn = *(const v2f*)(pin + k0);
        v2f cr = *(const v2f*)(phr + k0);
        v2f cz = *(const v2f*)(phz + k0);
        v2f cn = *(const v2f*)(phn + k0);
        air = wmma_f32(ax, br, air);
        aiz = wmma_f32(ax, bz, aiz);
        ain = wmma_f32(ax, bn, ain);
        ahr = wmma_f32(ah, cr, ahr);
        ahz = wmma_f32(ah, cz, ahz);
        ahn = wmma_f32(ah, cn, ahn);
    }

    // Per-lane column index in the C/D layout: n = lane & 15 (both halves).
    const int nc = c + m16;
    const float b_r  = bih[nc]       + bhh[nc];
    const float b_z  = bih[64 + nc]  + bhh[64 + nc];
    const float b_in = bih[128 + nc];
    const float b_hn = bhh[128 + nc];

#pragma unroll
    for (int i = 0; i < 8; ++i) {
        const int m = row0 + i + 8 * kh;
        const float hprev = h[(long long)m * D + nc];
        const float rg = sigmoid_f(air[i] + ahr[i] + b_r);
        const float zg = sigmoid_f(aiz[i] + ahz[i] + b_z);
        const float ng = tanhf((ain[i] + b_in) + rg * (ahn[i] + b_hn));
        out[(long long)m * D + nc] = (1.0f - zg) * ng + zg * hprev;
    }
}

// ---------------------------------------------------------------------------
// Host launcher.  Input order: h, edge_weights, W_ih, W_hh, b_ih, b_hh, src, dst
// ---------------------------------------------------------------------------
extern "C" void kernel_launch(void* const* d_in, const int* in_sizes, int n_in,
                              void* d_out, int out_size, void* d_ws, size_t ws_size,
                              hipStream_t stream) {
    const float* h   = (const float*)d_in[0];
    const float* ew  = (const float*)d_in[1];
    const float* Wih = (const float*)d_in[2];
    const float* Whh = (const float*)d_in[3];
    const float* bih = (const float*)d_in[4];
    const float* bhh = (const float*)d_in[5];
    const int*   src = (const int*)d_in[6];
    const int*   dst = (const int*)d_in[7];
    float* out  = (float*)d_out;
    float* hnew = (float*)d_ws;   // N_NODES*D floats = 25.6 MB scratch

    // 1) zero accumulator
    const int n4 = N_NODES * D / 4;
    zero_acc_kernel<<<(n4 + 255) / 256, 256, 0, stream>>>((float4*)hnew, n4);

    // 2) edge scatter: 16 threads per edge
    const long long nthr = (long long)N_EDGES * 16;
    edge_scatter_kernel<<<(int)((nthr + 255) / 256), 256, 0, stream>>>(
        h, ew, src, dst, hnew);

    // 3) fused WMMA GRU: 6250 row blocks x 4 feature blocks = 25000 waves,
    //    8 waves (256 threads) per block -> 3125 blocks exactly.
    const int nwaves = (N_NODES / 16) * 4;
    gru_wmma_kernel<<<nwaves / 8, 256, 0, stream>>>(
        hnew, h, Wih, Whh, bih, bhh, out);
}